// RefiningStrategy_7550552506781
// MI455X (gfx1250) — compile-verified
//
#include <hip/hip_runtime.h>

typedef __attribute__((ext_vector_type(2))) float v2f;
typedef __attribute__((ext_vector_type(4))) float v4f;
typedef __attribute__((ext_vector_type(8))) float v8f;

#define B_ 8
#define S_ 128
#define D_ 480
#define E_ 8
#define C_ 10
#define F_ 994   // 480 + 480 + 8 + 8 + 8 + 10
#define ROWS_ (B_ * S_)   // 1024

// ---------------------------------------------------------------------------
// Kernel 1: per-row separable terms via f32 WMMA (V_WMMA_F32_16X16X4_F32).
//   colterm[row][k] = sum_d node[row,d]*W[k,    d] + sum_e edge[row,e]*W[k,960+e]
//   rowterm[row][k] = sum_d node[row,d]*W[k,480+d] + sum_e edge[row,e]*W[k,968+e] + bias[k]
// where edge[row,e] = weight_prob[b,s,s,e], row = b*S+s.
// One wave per 16-row tile; K stepped by 4; two accumulators (N-tile col / row).
// A layout (16x4 f32): lanes 0-15 carry M=0..15 with K={k0,k0+1},
//                      lanes 16-31 carry M with K={k0+2,k0+3}.
// D layout: VGPR r -> M=r (lanes 0-15) / M=r+8 (lanes 16-31), N = lane&15.
//
// Lanes with n>=10 feed a clamped (in-bounds) W row into B; they only pollute
// D columns 10..15 which are never read — so no EXEC masking in the hot loop.
// ---------------------------------------------------------------------------
__global__ __launch_bounds__(32) void precompute_terms(
    const float* __restrict__ node,   // [B*S, 480]
    const float* __restrict__ wp,     // [B,S,S,E]
    const float* __restrict__ W,      // [10, 994]
    const float* __restrict__ bias,   // [10]
    float* __restrict__ colterm,      // [B*S, 16] (only k<10 valid)
    float* __restrict__ rowterm)      // [B*S, 16]
{
    const int tile = blockIdx.x;        // 0..63
    const int lane = threadIdx.x;       // 0..31 (wave32)
    const int n    = lane & 15;         // N (output channel) this lane holds
    const int half = lane >> 4;         // 0: K 0,1 / M 0..7 ; 1: K 2,3 / M 8..15
    const int arow = tile * 16 + n;     // A-matrix row fed by this lane
    const bool nval = (n < C_);
    const int nc   = nval ? n : (C_ - 1);   // clamped: keeps all loads in-bounds

    const float* __restrict__ ap = node + (size_t)arow * D_ + 2 * half;
    const float* __restrict__ wn = W + (size_t)nc * F_ + 2 * half;

    v8f acc0 = {};   // col half: W[:, 0:480]
    v8f acc1 = {};   // row half: W[:, 480:960]
    #pragma unroll 4
    for (int k0 = 0; k0 < D_; k0 += 4) {
        const v2f a  = *(const v2f*)(ap + k0);
        const v2f b0 = *(const v2f*)(wn + k0);
        const v2f b1 = *(const v2f*)(wn + 480 + k0);
        // 8 args: (neg_a, A, neg_b, B, c_mod, C, reuse_a, reuse_b)
        acc0 = __builtin_amdgcn_wmma_f32_16x16x4_f32(
            false, a, false, b0, (short)0, acc0, false, false);
        acc1 = __builtin_amdgcn_wmma_f32_16x16x4_f32(
            false, a, false, b1, (short)0, acc1, false, false);
    }

    if (nval) {
        float w_ei[E_], w_ej[E_];
        #pragma unroll
        for (int e = 0; e < E_; ++e) {
            w_ei[e] = W[(size_t)n * F_ + 960 + e];   // edge_i weights (j-dep -> colterm)
            w_ej[e] = W[(size_t)n * F_ + 968 + e];   // edge_j weights (i-dep -> rowterm)
        }
        const float bn = bias[n];
        #pragma unroll
        for (int r = 0; r < 8; ++r) {
            const int row = tile * 16 + r + 8 * half;   // D-matrix M index
            const int s   = row & (S_ - 1);
            const float* __restrict__ edge = wp + ((size_t)row * S_ + s) * E_;
            float c0 = acc0[r];
            float c1 = acc1[r] + bn;
            #pragma unroll
            for (int e = 0; e < E_; ++e) {
                const float ev = edge[e];
                c0 += ev * w_ei[e];
                c1 += ev * w_ej[e];
            }
            colterm[(size_t)row * 16 + n] = c0;
            rowterm[(size_t)row * 16 + n] = c1;
        }
    }
}

// ---------------------------------------------------------------------------
// Kernel 2: bandwidth-bound streaming pass. One thread per (b,i,j):
//   out[b,i,j,k] = colterm[b,j,k] + rowterm[b,i,k]
//                + wp[b,i,j,:].W[k,976:984] + masked_label[b,i,j,:].W[k,984:994]
// Channel masks: k==0 untouched, k in 1..3 need j>=i (triu), k in 4..9 need j<=i.
// wp load: 32B/thread (2x float4). label: 40B (5x float2). store: 40B contiguous.
// Pairwise weights (10x18 floats) cached in LDS per block.
// ---------------------------------------------------------------------------
__global__ __launch_bounds__(256) void fuse_pairwise(
    const float* __restrict__ label,    // [B,S,S,10]
    const float* __restrict__ wp,       // [B,S,S,8]
    const float* __restrict__ W,        // [10, 994]
    const float* __restrict__ colterm,  // [B*S, 16]
    const float* __restrict__ rowterm,  // [B*S, 16]
    float* __restrict__ out)            // [B,S,S,10]
{
    __shared__ float sW[C_][E_ + C_];   // [k][0..7]=wp weights, [k][8..17]=label weights
    const int t = threadIdx.x;
    if (t < C_ * (E_ + C_)) {
        const int k = t / (E_ + C_);
        const int f = t % (E_ + C_);
        sW[k][f] = W[(size_t)k * F_ + 976 + f];
    }
    __syncthreads();

    const size_t tid = (size_t)blockIdx.x * blockDim.x + t;   // 0 .. B*S*S-1
    const int j  = (int)(tid & (S_ - 1));
    const int bi = (int)(tid >> 7);          // b*S + i
    const int i  = bi & (S_ - 1);
    const int bS = bi >> 7;                  // b

    // weight_prob[b,i,j,0:8]
    const float* __restrict__ wpp = wp + tid * E_;
    const v4f wp0 = *(const v4f*)(wpp);
    const v4f wp1 = *(const v4f*)(wpp + 4);
    float wv[E_] = { wp0.x, wp0.y, wp0.z, wp0.w, wp1.x, wp1.y, wp1.z, wp1.w };

    // label_sm[b,i,j,0:10] with triangular channel masks
    const float* __restrict__ lbp = label + tid * C_;
    float lm[C_];
    #pragma unroll
    for (int c = 0; c < C_; c += 2) {
        const v2f lv = *(const v2f*)(lbp + c);
        lm[c] = lv.x; lm[c + 1] = lv.y;
    }
    const float mu = (j >= i) ? 1.0f : 0.0f;   // triu: channels 1..3
    const float ml = (j <= i) ? 1.0f : 0.0f;   // tril: channels 4..9
    #pragma unroll
    for (int c = 1; c < 4; ++c) lm[c] *= mu;
    #pragma unroll
    for (int c = 4; c < C_; ++c) lm[c] *= ml;

    const float* __restrict__ ct = colterm + ((size_t)(bS * S_ + j)) * 16;
    const float* __restrict__ rt = rowterm + ((size_t)bi) * 16;

    float o[C_];
    #pragma unroll
    for (int k = 0; k < C_; ++k) {
        float acc = ct[k] + rt[k];
        #pragma unroll
        for (int e = 0; e < E_; ++e) acc += wv[e] * sW[k][e];
        #pragma unroll
        for (int c = 0; c < C_; ++c) acc += lm[c] * sW[k][E_ + c];
        o[k] = acc;
    }

    float* __restrict__ op = out + tid * C_;
    #pragma unroll
    for (int k = 0; k < C_; k += 2) {
        v2f ov; ov.x = o[k]; ov.y = o[k + 1];
        *(v2f*)(op + k) = ov;
    }
}

extern "C" void kernel_launch(void* const* d_in, const int* in_sizes, int n_in,
                              void* d_out, int out_size, void* d_ws, size_t ws_size,
                              hipStream_t stream) {
    (void)in_sizes; (void)n_in; (void)out_size; (void)ws_size;
    const float* node  = (const float*)d_in[0];   // gcn_outputs [8,128,480]
    const float* label = (const float*)d_in[1];   // label_sm   [8,128,128,10]
    const float* wp    = (const float*)d_in[2];   // weight_prob[8,128,128,8]
    // d_in[3] = tensor_masks: unused by the reference computation
    const float* W     = (const float*)d_in[4];   // W_weight [10,994]
    const float* bias  = (const float*)d_in[5];   // W_bias   [10]
    float* out = (float*)d_out;

    float* colterm = (float*)d_ws;                   // 1024*16 floats = 64 KB
    float* rowterm = colterm + (size_t)ROWS_ * 16;   // next 64 KB

    precompute_terms<<<ROWS_ / 16, 32, 0, stream>>>(node, wp, W, bias, colterm, rowterm);
    fuse_pairwise<<<(B_ * S_ * S_) / 256, 256, 0, stream>>>(label, wp, W, colterm, rowterm, out);
}